// DynamicRetriever_15221364097574
// MI455X (gfx1250) — compile-verified
//
#include <hip/hip_runtime.h>
#include <math.h>

#define N_ROWS 2048
#define DIM    512
#define VOCAB  32000
#define TOPK   16

typedef float v2f __attribute__((ext_vector_type(2)));
typedef float v8f __attribute__((ext_vector_type(8)));

__device__ __forceinline__ float sigmoidf_(float x) {
    return 1.0f / (1.0f + __expf(-x));
}

// ---------------------------------------------------------------------------
// Kernel 1: per row — mean(searched_hidden), bandwidth, sparse softmax, merged
// block = 256 threads, grid = N_ROWS. searched_hidden read exactly once
// (K values per d kept in registers across the softmax barrier).
// ---------------------------------------------------------------------------
__global__ void dr_row_stats(const float* __restrict__ hidden,
                             const float* __restrict__ distances,
                             const float* __restrict__ searched_hidden,
                             const float* __restrict__ bw_W,
                             const float* __restrict__ bw_b,
                             float* __restrict__ merged_out,
                             float* __restrict__ sparse_out) {
    const int n   = blockIdx.x;
    const int tid = threadIdx.x;
    const int d0  = tid;          // first feature this thread owns
    const int d1  = tid + 256;    // second feature

    __shared__ float red[256];
    __shared__ float spd[TOPK];
    __shared__ float bw_sh;

    const float* sh_base = searched_hidden + (size_t)n * TOPK * DIM;

    float s0[TOPK], s1[TOPK];
    float sum0 = 0.0f, sum1 = 0.0f;
#pragma unroll
    for (int k = 0; k < TOPK; ++k) {
        s0[k] = sh_base[k * DIM + d0];
        s1[k] = sh_base[k * DIM + d1];
        sum0 += s0[k];
        sum1 += s1[k];
    }
    const float mean0 = sum0 * (1.0f / TOPK);
    const float mean1 = sum1 * (1.0f / TOPK);

    // partial dot for bandwidth linear: [hidden | mean] . bw_W
    float p = hidden[n * DIM + d0] * bw_W[d0]
            + hidden[n * DIM + d1] * bw_W[d1]
            + mean0 * bw_W[DIM + d0]
            + mean1 * bw_W[DIM + d1];
    red[tid] = p;
    __syncthreads();
    for (int s = 128; s > 0; s >>= 1) {
        if (tid < s) red[tid] += red[tid + s];
        __syncthreads();
    }
    if (tid == 0) {
        bw_sh = sigmoidf_(red[0] + bw_b[0]);
    }
    __syncthreads();

    if (tid == 0) {
        const float inv_bw = 1.0f / bw_sh;
        float sc[TOPK];
        float mx = -INFINITY;
#pragma unroll
        for (int k = 0; k < TOPK; ++k) {
            sc[k] = -distances[n * TOPK + k] * inv_bw;
            mx = fmaxf(mx, sc[k]);
        }
        float se = 0.0f;
#pragma unroll
        for (int k = 0; k < TOPK; ++k) { sc[k] = __expf(sc[k] - mx); se += sc[k]; }
        const float inv = 1.0f / se;
#pragma unroll
        for (int k = 0; k < TOPK; ++k) {
            const float pk = sc[k] * inv;
            spd[k] = pk;
            sparse_out[n * TOPK + k] = pk;
        }
    }
    __syncthreads();

    float m0 = 0.0f, m1 = 0.0f;
#pragma unroll
    for (int k = 0; k < TOPK; ++k) {
        m0 += spd[k] * s0[k];
        m1 += spd[k] * s1[k];
    }
    merged_out[(size_t)n * DIM + d0] = m0;
    merged_out[(size_t)n * DIM + d1] = m1;
}

// ---------------------------------------------------------------------------
// Kernel 2: mixing = sigmoid(relu([hidden|merged]@W1 + b1) @ W2 + b2)
// FP32 WMMA (V_WMMA_F32_16X16X4_F32). Block handles 16 batch rows; 8 waves,
// each wave owns 4 16x16 output tiles (8*4*16 = 512 columns). A staged in LDS
// (pitch 516 to dodge bank conflicts); h1 never materialized — fused into an
// LDS per-row dot with W2.
// ---------------------------------------------------------------------------
#define A_PITCH 516

__global__ void dr_mixing_wmma(const float* __restrict__ hidden,
                               const float* __restrict__ merged,
                               const float* __restrict__ mw_W1,
                               const float* __restrict__ mw_b1,
                               const float* __restrict__ mw_W2,
                               const float* __restrict__ mw_b2,
                               float* __restrict__ mixing_out) {
    const int tid  = threadIdx.x;
    const int wid  = tid >> 5;            // wave id, 0..7
    const int lane = tid & 31;
    const int half = lane >> 4;           // 0: K even pair, 1: K odd pair
    const int mrow = lane & 15;           // M row (for A) / N col (for B)
    const int row0 = blockIdx.x * 16;

    __shared__ float As[16][A_PITCH];
    __shared__ float rowsum[16];

    if (tid < 16) rowsum[tid] = 0.0f;

    const v8f zero = {0.f, 0.f, 0.f, 0.f, 0.f, 0.f, 0.f, 0.f};
    v8f acc[4];
#pragma unroll
    for (int t = 0; t < 4; ++t) acc[t] = zero;

    for (int phase = 0; phase < 2; ++phase) {
        __syncthreads();
        const float* src = (phase == 0) ? hidden : merged;
        // cooperative load of 16 x 512 A tile
        for (int li = tid; li < 16 * DIM; li += 256) {
            const int r = li >> 9;
            const int c = li & (DIM - 1);
            As[r][c] = src[(size_t)(row0 + r) * DIM + c];
        }
        __syncthreads();

        const float* Wb = mw_W1 + (size_t)(phase * DIM) * DIM;
        for (int kk = 0; kk < DIM; kk += 4) {
            const int kr = kk + 2 * half;
            v2f a;
            a.x = As[mrow][kr];
            a.y = As[mrow][kr + 1];
#pragma unroll
            for (int t = 0; t < 4; ++t) {
                const int col = (wid * 4 + t) * 16 + mrow;
                v2f b;
                b.x = Wb[(size_t)kr * DIM + col];
                b.y = Wb[(size_t)(kr + 1) * DIM + col];
                acc[t] = __builtin_amdgcn_wmma_f32_16x16x4_f32(
                    false, a, false, b, (short)0, acc[t], false, false);
            }
        }
    }
    __syncthreads();

    // bias + relu + fold into h1 . W2 per batch row
#pragma unroll
    for (int t = 0; t < 4; ++t) {
        const int col = (wid * 4 + t) * 16 + mrow;
        const float b1 = mw_b1[col];
        const float w2 = mw_W2[col];
#pragma unroll
        for (int i = 0; i < 8; ++i) {
            const int m = i + 8 * half;
            float v = acc[t][i] + b1;
            v = fmaxf(v, 0.0f);
            atomicAdd(&rowsum[m], v * w2);
        }
    }
    __syncthreads();

    if (tid < 16) {
        mixing_out[row0 + tid] = sigmoidf_(rowsum[tid] + mw_b2[0]);
    }
}

// ---------------------------------------------------------------------------
// Kernel 3: online (max, sumexp) over logits per row. Single pass, float4.
// ---------------------------------------------------------------------------
__global__ void dr_logit_stats(const float* __restrict__ logits,
                               float* __restrict__ row_max,
                               float* __restrict__ row_sum) {
    const int n   = blockIdx.x;
    const int tid = threadIdx.x;
    const float4* lg = (const float4*)(logits + (size_t)n * VOCAB);

    float m = -INFINITY, s = 0.0f;
    for (int j = tid; j < VOCAB / 4; j += 256) {
        const float4 x = lg[j];
        const float xs[4] = {x.x, x.y, x.z, x.w};
#pragma unroll
        for (int c = 0; c < 4; ++c) {
            const float v = xs[c];
            if (v > m) { s = s * __expf(m - v) + 1.0f; m = v; }
            else       { s += __expf(v - m); }
        }
    }

    __shared__ float ms[256];
    __shared__ float ss[256];
    ms[tid] = m; ss[tid] = s;
    __syncthreads();
    for (int st = 128; st > 0; st >>= 1) {
        if (tid < st) {
            const float m1 = ms[tid], s1 = ss[tid];
            const float m2 = ms[tid + st], s2 = ss[tid + st];
            const float mm = fmaxf(m1, m2);
            ms[tid] = mm;
            ss[tid] = s1 * __expf(m1 - mm) + s2 * __expf(m2 - mm);
        }
        __syncthreads();
    }
    if (tid == 0) { row_max[n] = ms[0]; row_sum[n] = ss[0]; }
}

// ---------------------------------------------------------------------------
// Kernel 4: out = log((1-m)*softmax(logits) + m*scatter(sparse)).
// Fast path (no scatter hit): x - max + log((1-m)/sum)  — no exp/log.
// ---------------------------------------------------------------------------
__global__ void dr_output(const float* __restrict__ logits,
                          const int* __restrict__ token_indices,
                          const float* __restrict__ sparse,
                          const float* __restrict__ mixing,
                          const float* __restrict__ row_max,
                          const float* __restrict__ row_sum,
                          float* __restrict__ out) {
    const int n   = blockIdx.x;
    const int tid = threadIdx.x;

    __shared__ int   idx_s[TOPK];
    __shared__ float msp_s[TOPK];

    const float mix  = mixing[n];
    const float rmax = row_max[n];
    const float coef = (1.0f - mix) / row_sum[n];
    const float logc = logf(coef);

    if (tid < TOPK) {
        idx_s[tid] = token_indices[n * TOPK + tid];
        msp_s[tid] = mix * sparse[n * TOPK + tid];
    }
    __syncthreads();

    const float4* lg = (const float4*)(logits + (size_t)n * VOCAB);
    float4*       ot = (float4*)(out + (size_t)n * VOCAB);

    for (int j4 = tid; j4 < VOCAB / 4; j4 += 256) {
        const float4 x = lg[j4];
        const float xs[4] = {x.x, x.y, x.z, x.w};
        float rs[4];
#pragma unroll
        for (int c = 0; c < 4; ++c) {
            const int j = 4 * j4 + c;
            float add = 0.0f;
#pragma unroll
            for (int k = 0; k < TOPK; ++k) {
                add += (idx_s[k] == j) ? msp_s[k] : 0.0f;
            }
            const float xm = xs[c] - rmax;
            rs[c] = (add > 0.0f) ? logf(coef * __expf(xm) + add)
                                 : (xm + logc);
        }
        float4 r; r.x = rs[0]; r.y = rs[1]; r.z = rs[2]; r.w = rs[3];
        ot[j4] = r;
    }
}

// ---------------------------------------------------------------------------
extern "C" void kernel_launch(void* const* d_in, const int* in_sizes, int n_in,
                              void* d_out, int out_size, void* d_ws, size_t ws_size,
                              hipStream_t stream) {
    const float* hidden          = (const float*)d_in[0];
    const float* logits          = (const float*)d_in[1];
    const float* distances       = (const float*)d_in[2];
    const float* searched_hidden = (const float*)d_in[3];
    const int*   token_indices   = (const int*)d_in[4];
    const float* bw_W            = (const float*)d_in[5];
    const float* bw_b            = (const float*)d_in[6];
    const float* mw_W1           = (const float*)d_in[7];
    const float* mw_b1           = (const float*)d_in[8];
    const float* mw_W2           = (const float*)d_in[9];
    const float* mw_b2           = (const float*)d_in[10];
    float* out = (float*)d_out;

    // workspace layout (floats)
    float* ws      = (float*)d_ws;
    float* merged  = ws;                              // N*D
    float* sparse  = merged + (size_t)N_ROWS * DIM;   // N*K
    float* mixing  = sparse + (size_t)N_ROWS * TOPK;  // N
    float* row_max = mixing + N_ROWS;                 // N
    float* row_sum = row_max + N_ROWS;                // N

    dr_row_stats<<<N_ROWS, 256, 0, stream>>>(hidden, distances, searched_hidden,
                                             bw_W, bw_b, merged, sparse);

    dr_mixing_wmma<<<N_ROWS / 16, 256, 0, stream>>>(hidden, merged, mw_W1, mw_b1,
                                                    mw_W2, mw_b2, mixing);

    dr_logit_stats<<<N_ROWS, 256, 0, stream>>>(logits, row_max, row_sum);

    dr_output<<<N_ROWS, 256, 0, stream>>>(logits, token_indices, sparse, mixing,
                                          row_max, row_sum, out);
}